// MPNNNet_78211354460179
// MI455X (gfx1250) — compile-verified
//
#include <hip/hip_runtime.h>

typedef __attribute__((ext_vector_type(2))) float v2f;
typedef __attribute__((ext_vector_type(8))) float v8f;

#define NN      10000
#define EE      40000
#define HID     64
#define NGRAPH  512
#define BN_EPS  1e-5f

__device__ __forceinline__ v8f wmma4(v2f a, v2f b, v8f c) {
    return __builtin_amdgcn_wmma_f32_16x16x4_f32(false, a, false, b, (short)0, c, false, false);
}

// ---------------------------------------------------------------------------
// utility kernels
// ---------------------------------------------------------------------------
__global__ void zero_kernel(float* __restrict__ p, int n) {
    int i = blockIdx.x * blockDim.x + threadIdx.x;
    if (i < n) p[i] = 0.f;
}

// Pack w2/b2 into WMMA-fragment-major B panel.
// Logical K-row t = i*64 + k  (i = input channel, k = edge-hidden channel):
//   t <  KE          : W2p[t,c] = w2[k, i*64+c]
//   KE <= t < KE+in  : W2p[t,c] = b2[(t-KE)*64+c]   (bias rides inside GEMM)
//   else             : 0 (pad to KT = multiple of 4)
// Fragment-major storage: w2q[(( (t>>2)*2 + half )*64 + c)*2 + j] = W2p[4*(t>>2)+2*half+j, c]
// so one lane's two K-values for a 16x16x4 WMMA B fragment are one b64 load.
__global__ void prep_w2q_kernel(const float* __restrict__ w2,
                                const float* __restrict__ b2,
                                float* __restrict__ w2q,
                                int in_ch, int KT) {
    int idx = blockIdx.x * blockDim.x + threadIdx.x;
    if (idx >= KT * HID) return;
    int j    = idx & 1;
    int c    = (idx >> 1) & 63;
    int gh   = idx >> 7;          // g*2 + half
    int half = gh & 1;
    int g    = gh >> 1;
    int t    = g * 4 + half * 2 + j;
    int KE   = in_ch * HID;
    float v = 0.f;
    if (t < KE) {
        int i = t >> 6, k = t & 63;
        v = w2[(size_t)k * (in_ch * HID) + i * HID + c];
    } else if (t < KE + in_ch) {
        v = b2[(t - KE) * HID + c];
    }
    w2q[idx] = v;
}

// ---------------------------------------------------------------------------
// Fused NNConv message GEMM:  msg[e,:] = P[e,:] @ W2p,  P[e, i*64+k] = h[e,k]*xs[e,i]
// Block = 64 threads (2 waves), M-tile = 32 edges, each wave owns 32 output cols.
// Per wave / K-step: 4 fp32 WMMA (2 M-tiles x 2 N-tiles) sharing one B row-pair.
// Branch-free main loop; xs[i] hoisted (constant over 16 consecutive K-steps).
// Epilogue: atomic scatter-add into aggr[dst] + edge counts for segment-mean.
// ---------------------------------------------------------------------------
template <int IN_CH>
__global__ void __launch_bounds__(64)
nnconv_gemm_kernel(const float* __restrict__ x,      // [N, IN_CH]
                   const int*   __restrict__ src,    // [E]
                   const int*   __restrict__ dst,    // [E]
                   const float* __restrict__ ea,     // [E, 3]
                   const float* __restrict__ w1,     // [3, 64]
                   const float* __restrict__ b1,     // [64]
                   const float* __restrict__ w2q,    // packed [KT*64]
                   float* __restrict__ aggr,         // [N, 64]  (atomic)
                   float* __restrict__ cnt)          // [N]      (atomic)
{
    constexpr int KE = IN_CH * HID;                 // multiple of 4 for 11 and 64
    constexpr int KT = (KE + IN_CH + 3) & ~3;

    __shared__ float h_lds [32 * HID];              // relu(ea@w1+b1), 32 edges
    __shared__ float xs_lds[32 * HID];              // gathered x[src]
    __shared__ int   dst_lds[32];

    const int tid  = threadIdx.x;                   // 0..63 (2 waves)
    const int lane = tid & 31;
    const int nh   = tid >> 5;                      // wave id = N-half (cols nh*32..)
    const int half = lane >> 4;                     // intra-wave half
    const int lm   = lane & 15;
    const int e0   = blockIdx.x * 32;

    if (tid < 32) dst_lds[tid] = dst[e0 + tid];

    // edge MLP: h[m,c] = relu(ea[e]*w1[:,c] + b1[c]); 32x64 values over 64 threads
    for (int t = tid; t < 32 * HID; t += 64) {
        int m = t >> 6, c = t & 63;
        const float* eap = ea + (size_t)(e0 + m) * 3;
        float v = eap[0] * w1[c] + eap[1] * w1[64 + c] + eap[2] * w1[128 + c] + b1[c];
        h_lds[t] = v > 0.f ? v : 0.f;
    }
    // gather source-node features (row stride 64 in LDS regardless of IN_CH)
    for (int t = tid; t < 32 * IN_CH; t += 64) {
        int m = t / IN_CH, i = t - m * IN_CH;       // IN_CH is constexpr -> cheap
        xs_lds[m * HID + i] = x[(size_t)src[e0 + m] * IN_CH + i];
    }
    __syncthreads();

    v8f acc00 = {}, acc01 = {}, acc10 = {}, acc11 = {};
    const float* hA = h_lds + lm * HID;             // A rows lm and lm+16
    const float* hB = h_lds + (lm + 16) * HID;
    const int cbase = (nh * 32 + lm) * 2;           // packed col offset for this lane

    // ---- main loop: outer-product region, no boundary checks ----
    for (int i = 0; i < IN_CH; ++i) {
        const float xvA = xs_lds[lm * HID + i];
        const float xvB = xs_lds[(lm + 16) * HID + i];
        const float* bp = w2q + (size_t)((i * 16) * 2 + half) * 128 + cbase;
        #pragma unroll 4
        for (int kc = 0; kc < HID; kc += 4) {
            v2f B0 = *(const v2f*)(bp);
            v2f B1 = *(const v2f*)(bp + 32);
            bp += 256;                              // next group g (2 halves * 128)
            const int kb = kc + 2 * half;
            v2f A0 = *(const v2f*)(hA + kb) * xvA;  // ds_load_b64 + packed mul
            v2f A1 = *(const v2f*)(hB + kb) * xvB;
            acc00 = wmma4(A0, B0, acc00);
            acc01 = wmma4(A0, B1, acc01);
            acc10 = wmma4(A1, B0, acc10);
            acc11 = wmma4(A1, B1, acc11);
        }
    }

    // ---- bias tail: K-rows [KE, KT), P = xs (masked), branch-free selects ----
    for (int g = KE >> 2; g < (KT >> 2); ++g) {
        const float* bp = w2q + (size_t)(g * 2 + half) * 128 + cbase;
        v2f B0 = *(const v2f*)(bp);
        v2f B1 = *(const v2f*)(bp + 32);
        const int t0 = g * 4 + 2 * half;
        v2f A0, A1;
        #pragma unroll
        for (int j = 0; j < 2; ++j) {
            int ib  = t0 + j - KE;
            int ibc = ib < IN_CH ? ib : 0;
            float msk = ib < IN_CH ? 1.f : 0.f;
            A0[j] = xs_lds[lm * HID + ibc] * msk;
            A1[j] = xs_lds[(lm + 16) * HID + ibc] * msk;
        }
        acc00 = wmma4(A0, B0, acc00);
        acc01 = wmma4(A0, B1, acc01);
        acc10 = wmma4(A1, B0, acc10);
        acc11 = wmma4(A1, B1, acc11);
    }

    // D layout: VGPR r, lanes 0-15: M=r, N=lane; lanes 16-31: M=r+8, N=lane-16
    #pragma unroll
    for (int r = 0; r < 8; ++r) {
        const int m = r + 8 * half;
        size_t b0 = (size_t)dst_lds[m]      * HID + nh * 32 + lm;
        size_t b1 = (size_t)dst_lds[m + 16] * HID + nh * 32 + lm;
        atomicAdd(&aggr[b0],      acc00[r]);
        atomicAdd(&aggr[b0 + 16], acc01[r]);
        atomicAdd(&aggr[b1],      acc10[r]);
        atomicAdd(&aggr[b1 + 16], acc11[r]);
    }
    if (tid < 32) atomicAdd(&cnt[dst_lds[tid]], 1.f);
}

// ---------------------------------------------------------------------------
// out[v,o] = aggr[v,o]/max(cnt,1) + x[v,:]@root[:,o] + bias[o]
// ---------------------------------------------------------------------------
__global__ void root_update_kernel(const float* __restrict__ aggr,
                                   const float* __restrict__ cnt,
                                   const float* __restrict__ x,
                                   const float* __restrict__ root,
                                   const float* __restrict__ bias,
                                   float* __restrict__ out, int in_ch) {
    int idx = blockIdx.x * blockDim.x + threadIdx.x;
    if (idx >= NN * HID) return;
    int v = idx >> 6, o = idx & 63;
    float c = cnt[v];
    float s = aggr[idx] / (c > 1.f ? c : 1.f);
    float r = bias[o];
    const float* xr = x + (size_t)v * in_ch;
    for (int i = 0; i < in_ch; ++i) r += xr[i] * root[i * HID + o];
    out[idx] = s + r;
}

// per-channel mean / biased variance over N nodes
__global__ void bn_stats_kernel(const float* __restrict__ xt,
                                float* __restrict__ mu, float* __restrict__ var) {
    __shared__ float s1[256], s2[256];
    int c = blockIdx.x, t = threadIdx.x;
    float a = 0.f, b = 0.f;
    for (int v = t; v < NN; v += 256) {
        float q = xt[(size_t)v * HID + c];
        a += q; b += q * q;
    }
    s1[t] = a; s2[t] = b; __syncthreads();
    for (int s = 128; s > 0; s >>= 1) {
        if (t < s) { s1[t] += s1[t + s]; s2[t] += s2[t + s]; }
        __syncthreads();
    }
    if (t == 0) {
        float m = s1[0] / (float)NN;
        mu[c]  = m;
        var[c] = s2[0] / (float)NN - m * m;
    }
}

__global__ void bn_apply_kernel(const float* __restrict__ xt,
                                const float* __restrict__ mu,
                                const float* __restrict__ var,
                                const float* __restrict__ gamma,
                                const float* __restrict__ beta,
                                const float* __restrict__ xres,  // nullptr => no residual
                                float* __restrict__ out) {
    int idx = blockIdx.x * blockDim.x + threadIdx.x;
    if (idx >= NN * HID) return;
    int o = idx & 63;
    float y = (xt[idx] - mu[o]) * rsqrtf(var[o] + BN_EPS) * gamma[o] + beta[o];
    y = y > 0.f ? y : 0.f;
    if (xres) y += xres[idx];
    out[idx] = y;
}

// global mean pool (scatter-add + counts)
__global__ void pool_kernel(const float* __restrict__ x,
                            const int* __restrict__ batch,
                            float* __restrict__ pooled, float* __restrict__ gcnt) {
    int idx = blockIdx.x * blockDim.x + threadIdx.x;
    if (idx >= NN * HID) return;
    int v = idx >> 6, o = idx & 63;
    int g = batch[v];
    atomicAdd(&pooled[(size_t)g * HID + o], x[idx]);
    if (o == 0) atomicAdd(&gcnt[g], 1.f);
}

// per-graph MLP 64 -> 64 -> 64 -> 1
__global__ void mlp_kernel(const float* __restrict__ pooled,
                           const float* __restrict__ gcnt,
                           const float* __restrict__ w1, const float* __restrict__ b1,
                           const float* __restrict__ w2, const float* __restrict__ b2,
                           const float* __restrict__ w3, const float* __restrict__ b3,
                           float* __restrict__ out) {
    __shared__ float g[HID], h1[HID], h2[HID];
    int b = blockIdx.x, o = threadIdx.x;
    float c = gcnt[b]; c = c > 1.f ? c : 1.f;
    g[o] = pooled[(size_t)b * HID + o] / c;
    __syncthreads();
    float s = b1[o];
    for (int i = 0; i < HID; ++i) s += g[i] * w1[i * HID + o];
    h1[o] = s > 0.f ? s : 0.f;
    __syncthreads();
    s = b2[o];
    for (int i = 0; i < HID; ++i) s += h1[i] * w2[i * HID + o];
    h2[o] = s > 0.f ? s : 0.f;
    __syncthreads();
    if (o == 0) {
        float t = b3[0];
        for (int i = 0; i < HID; ++i) t += h2[i] * w3[i];
        out[b] = t;
    }
}

// ---------------------------------------------------------------------------
extern "C" void kernel_launch(void* const* d_in, const int* in_sizes, int n_in,
                              void* d_out, int out_size, void* d_ws, size_t ws_size,
                              hipStream_t stream) {
    (void)in_sizes; (void)n_in; (void)out_size; (void)ws_size;

    const float* x0    = (const float*)d_in[0];          // [N, 11]
    const int*   ei    = (const int*)  d_in[1];          // [2, E]
    const int*   batch = (const int*)  d_in[2];          // [N]
    const float* ea    = (const float*)d_in[3];          // [E, 3]
    const int* src = ei;
    const int* dst = ei + EE;

    // workspace carve-out (~12 MB of fp32)
    float* ws = (float*)d_ws;
    size_t off = 0;
    float* aggr   = ws + off; off += (size_t)NN * HID;
    float* cnt    = ws + off; off += NN;
    float* tmp    = ws + off; off += (size_t)NN * HID;
    float* bufA   = ws + off; off += (size_t)NN * HID;
    float* bufB   = ws + off; off += (size_t)NN * HID;
    float* w2q    = ws + off; off += (size_t)(HID * HID + HID + 4) * HID;  // 4160*64 max
    float* mu     = ws + off; off += HID;
    float* var    = ws + off; off += HID;
    float* pooled = ws + off; off += (size_t)NGRAPH * HID;
    float* gcnt   = ws + off; off += NGRAPH;

    const int TPB = 256;
    const int NB_NH = (NN * HID + TPB - 1) / TPB;

    const float* xin = x0;
    float* xout = bufA;
    for (int l = 0; l < 4; ++l) {
        const int in_ch = (l == 0) ? 11 : HID;
        const int KT = (in_ch * HID + in_ch + 3) & ~3;
        const float* w1    = (const float*)d_in[4 + 8 * l + 0];
        const float* b1    = (const float*)d_in[4 + 8 * l + 1];
        const float* w2    = (const float*)d_in[4 + 8 * l + 2];
        const float* b2    = (const float*)d_in[4 + 8 * l + 3];
        const float* root  = (const float*)d_in[4 + 8 * l + 4];
        const float* bias  = (const float*)d_in[4 + 8 * l + 5];
        const float* gamma = (const float*)d_in[4 + 8 * l + 6];
        const float* beta  = (const float*)d_in[4 + 8 * l + 7];

        prep_w2q_kernel<<<(KT * HID + TPB - 1) / TPB, TPB, 0, stream>>>(w2, b2, w2q, in_ch, KT);
        zero_kernel<<<NB_NH, TPB, 0, stream>>>(aggr, NN * HID);
        zero_kernel<<<(NN + TPB - 1) / TPB, TPB, 0, stream>>>(cnt, NN);

        if (l == 0)
            nnconv_gemm_kernel<11><<<EE / 32, 64, 0, stream>>>(xin, src, dst, ea, w1, b1, w2q, aggr, cnt);
        else
            nnconv_gemm_kernel<64><<<EE / 32, 64, 0, stream>>>(xin, src, dst, ea, w1, b1, w2q, aggr, cnt);

        root_update_kernel<<<NB_NH, TPB, 0, stream>>>(aggr, cnt, xin, root, bias, tmp, in_ch);
        bn_stats_kernel<<<HID, 256, 0, stream>>>(tmp, mu, var);
        bn_apply_kernel<<<NB_NH, TPB, 0, stream>>>(tmp, mu, var, gamma, beta,
                                                   (l == 0) ? (const float*)nullptr : xin, xout);
        xin = xout;
        xout = (xout == bufA) ? bufB : bufA;
    }

    zero_kernel<<<(NGRAPH * HID + TPB - 1) / TPB, TPB, 0, stream>>>(pooled, NGRAPH * HID);
    zero_kernel<<<(NGRAPH + TPB - 1) / TPB, TPB, 0, stream>>>(gcnt, NGRAPH);
    pool_kernel<<<NB_NH, TPB, 0, stream>>>(xin, batch, pooled, gcnt);

    mlp_kernel<<<NGRAPH, HID, 0, stream>>>(pooled, gcnt,
                                           (const float*)d_in[36], (const float*)d_in[37],
                                           (const float*)d_in[38], (const float*)d_in[39],
                                           (const float*)d_in[40], (const float*)d_in[41],
                                           (float*)d_out);
}